// HierarchicalSampler_549755814572
// MI455X (gfx1250) — compile-verified
//
#include <hip/hip_runtime.h>
#include <math.h>

#define NC 64
#define NF 128
#define NT 192            // NC + NF
#define WPB 8             // waves (rays) per block
#define THREADS (WPB * 32)

#define MIND 2.0f
#define MAXD 6.0f

// gfx1250 async global->LDS staging (guarded so host pass / older clang still compiles)
#if defined(__AMDGCN__) &&                                             \
    __has_builtin(__builtin_amdgcn_global_load_async_to_lds_b64) &&    \
    __has_builtin(__builtin_amdgcn_global_load_async_to_lds_b128) &&   \
    __has_builtin(__builtin_amdgcn_s_wait_asynccnt)
#define USE_ASYNC_LDS 1
typedef int v2i __attribute__((ext_vector_type(2)));
typedef int v4i __attribute__((ext_vector_type(4)));
typedef __attribute__((address_space(1))) v2i Gv2i;   // global (AS1)
typedef __attribute__((address_space(3))) v2i Lv2i;   // LDS (AS3)
typedef __attribute__((address_space(1))) v4i Gv4i;
typedef __attribute__((address_space(3))) v4i Lv4i;
#else
#define USE_ASYNC_LDS 0
#endif

__device__ __forceinline__ void wave_fence() {
#if defined(__AMDGCN__)
  __builtin_amdgcn_wave_barrier();
#endif
}

// searchsorted(cdf[0..64], v, side='right') in [1,65]; branchless 7-step descent,
// loads always clamped in-bounds so no divergent branches are generated.
__device__ __forceinline__ int ub_cdf_65(const float* cdf, float v) {
  int p = 0;
#pragma unroll
  for (int s = 64; s >= 1; s >>= 1) {
    int t = p + s;
    float x = cdf[min(t - 1, NC)];
    bool c = (t <= NC + 1) && (x <= v);
    p = c ? t : p;
  }
  return p;
}

__global__ __launch_bounds__(THREADS) void hier_sampler_kernel(
    const float* __restrict__ origins, const float* __restrict__ dirs,
    const float* __restrict__ density, const float* __restrict__ trand,
    const float* __restrict__ uarr, float* __restrict__ out_pts,
    float* __restrict__ out_z, int nrays) {
  __shared__ float s_mrg[WPB][NT];      // [0,64): sorted coarse ; [64,192): sorted fine
  __shared__ float s_cdf[WPB][NC + 2];
  __shared__ float s_ze[WPB][NC + 2];
  __shared__ float s_in[WPB][NC + NF];  // density[64] then u[128]

  const int lane = threadIdx.x & 31;
  const int wv = threadIdx.x >> 5;
  const int ray = blockIdx.x * WPB + wv;
  if (ray >= nrays) return;  // whole-wave exit; no block barriers anywhere

  float* srt = s_mrg[wv];   // coarse run
  float* cdf = s_cdf[wv];
  float* ze = s_ze[wv];
  float* stg = s_in[wv];

  // ---- stage density + u into LDS (CDNA5 async data mover) ----
  const float* gden = density + (size_t)ray * NC;
  const float* gu = uarr + (size_t)ray * NF;
#if USE_ASYNC_LDS
  __builtin_amdgcn_global_load_async_to_lds_b64(
      (Gv2i*)(gden + 2 * lane), (Lv2i*)(stg + 2 * lane), 0, 0);
  __builtin_amdgcn_global_load_async_to_lds_b128(
      (Gv4i*)(gu + 4 * lane), (Lv4i*)(stg + NC + 4 * lane), 0, 0);
#else
  *(float2*)(stg + 2 * lane) = *(const float2*)(gden + 2 * lane);
  *(float4*)(stg + NC + 4 * lane) = *(const float4*)(gu + 4 * lane);
#endif

  // ---- ray origin/direction ----
  const float ox = origins[3 * ray], oy = origins[3 * ray + 1], oz = origins[3 * ray + 2];
  const float dx = dirs[3 * ray], dy = dirs[3 * ray + 1], dz = dirs[3 * ray + 2];

  // ---- coarse stratified z (2 consecutive samples per lane); monotone by construction ----
  const float step = (MAXD - MIND) / (float)(NC - 1);
  const float h = 0.5f * step;
  float2 tr = *(const float2*)(trand + (size_t)ray * NC + 2 * lane);
  const int e0 = 2 * lane, e1 = 2 * lane + 1;
  float z0 = MIND + step * (float)e0;
  float z1 = MIND + step * (float)e1;
  float lo0 = (e0 == 0) ? z0 : z0 - h;
  float up0 = z0 + h;                      // e0 even -> never index 63
  float lo1 = z1 - h;                      // e1 odd  -> never index 0
  float up1 = (e1 == NC - 1) ? z1 : z1 + h;
  float zv0 = lo0 + (up0 - lo0) * tr.x;
  float zv1 = lo1 + (up1 - lo1) * tr.y;
  *(float2*)(srt + e0) = make_float2(zv0, zv1);  // sorted coarse run in LDS [0,64)

  // deltas via shuffle (neighbor = next lane's zv0)
  float zv0n = __shfl_down(zv0, 1, 32);
  float d0 = zv1 - zv0;
  float d1 = (lane == 31) ? 1e10f : (zv0n - zv1);

#if USE_ASYNC_LDS
  __builtin_amdgcn_s_wait_asynccnt(0);
#endif
  wave_fence();
  float den0 = stg[e0], den1 = stg[e1];

  // alpha & exclusive cumprod (transmittance) via wave32 shuffle scan
  float a0 = 1.0f - __expf(-den0 * d0);
  float a1 = 1.0f - __expf(-den1 * d1);
  float c0 = 1.0f - a0 + 1e-10f;
  float c1 = 1.0f - a1 + 1e-10f;
  float sc = c0 * c1;
#pragma unroll
  for (int o = 1; o < 32; o <<= 1) {
    float t = __shfl_up(sc, o, 32);
    if (lane >= o) sc *= t;
  }
  float ex = __shfl_up(sc, 1, 32);
  if (lane == 0) ex = 1.0f;
  float w0 = a0 * ex + 1e-5f;            // weights + 1e-5
  float w1 = a1 * (ex * c0) + 1e-5f;

  // cdf: single scan of per-lane pair sums; total = lane31 of inclusive scan
  float ssw = w0 + w1;
#pragma unroll
  for (int o = 1; o < 32; o <<= 1) {
    float t = __shfl_up(ssw, o, 32);
    if (lane >= o) ssw += t;
  }
  float total = __shfl(ssw, 31, 32);
  float exw = __shfl_up(ssw, 1, 32);
  if (lane == 0) exw = 0.0f;
  float inv = 1.0f / total;
  if (lane == 0) cdf[0] = 0.0f;
  cdf[e0 + 1] = (exw + w0) * inv;
  cdf[e1 + 1] = (exw + w0 + w1) * inv;

  // z_edges[65]
  float zv1p = __shfl_up(zv1, 1, 32);
  ze[e0] = (lane == 0) ? (zv0 - 0.5f * (zv1 - zv0)) : 0.5f * (zv0 + zv1p);
  ze[e1] = 0.5f * (zv0 + zv1);
  if (lane == 31) ze[NC] = zv1 + 0.5f * (zv1 - zv0);
  wave_fence();

  // ---- inverse-CDF fine sampling (4 samples per lane), branchless searchsorted ----
  const float* ul = stg + NC;
  float r[4];
#pragma unroll
  for (int f = 0; f < 4; ++f) {
    float uv = ul[4 * lane + f];
    int ind = ub_cdf_65(cdf, uv);          // [1,65]
    int below = ind - 1;                   // <= 64
    int above = min(ind, NC);
    float cg0 = cdf[below], cg1 = cdf[above];
    float eg0 = ze[below], eg1 = ze[above];
    float dn = cg1 - cg0;
    dn = (dn < 1e-5f) ? 1.0f : dn;
    float t = (uv - cg0) / dn;
    r[f] = eg0 + t * (eg1 - eg0);
  }

  // ---- register bitonic sort of the 128 fine z (4 blocked values/lane: i = 4*lane+m) ----
  auto ce = [](float& a, float& b, bool up) {
    float mn = fminf(a, b), mx = fmaxf(a, b);
    a = up ? mn : mx;
    b = up ? mx : mn;
  };
  auto cex = [&](float& v, int q, bool up) {  // cross-lane compare-exchange via shfl_xor
    float p = __shfl_xor(v, q, 32);
    bool keepMin = up != ((lane & q) != 0);
    v = keepMin ? fminf(v, p) : fmaxf(v, p);
  };
  // k=2 (direction from element bit1)
  ce(r[0], r[1], true); ce(r[2], r[3], false);
  // k=4 (direction = bit2 of i = lane&1)
  {
    bool up = ((lane & 1) == 0);
    ce(r[0], r[2], up); ce(r[1], r[3], up);
    ce(r[0], r[1], up); ce(r[2], r[3], up);
  }
  // k=8..128: K4 = k/4; cross-lane merges then intra j=2,1
#pragma unroll
  for (int K4 = 2; K4 <= 32; K4 <<= 1) {
    bool up = ((lane & K4) == 0);          // K4==32 -> always ascending (final merge)
#pragma unroll
    for (int q = K4 >> 1; q >= 1; q >>= 1) {
#pragma unroll
      for (int m = 0; m < 4; ++m) cex(r[m], q, up);
    }
    ce(r[0], r[2], up); ce(r[1], r[3], up);
    ce(r[0], r[1], up); ce(r[2], r[3], up);
  }

  // sorted fine run -> LDS [64,192)
  float* sf = srt + NC;
  *(float4*)(sf + 4 * lane) = make_float4(r[0], r[1], r[2], r[3]);
  wave_fence();

  // ---- merge-path: lane owns outputs [6*lane, 6*lane+6) of the merged sequence ----
  const int dg = 6 * lane;
  // diagonal search: a = #coarse consumed before this diagonal (coarse wins ties).
  // Predicate srt[a] <= sf[dg-1-a] is monotone in a; 7-step branchless descent.
  int a;
  {
    const int loA = max(0, dg - NF), hiA = min(dg, NC);
    int p = loA;
#pragma unroll
    for (int s = 64; s >= 1; s >>= 1) {
      int t = p + s;
      float xc = srt[min(t - 1, NC - 1)];
      float xf = sf[max(min(dg - t, NF - 1), 0)];
      bool c = (t <= hiA) && (xc <= xf);
      p = c ? t : p;
    }
    a = p;
  }
  int b = dg - a;

  const float INF = __builtin_inff();
  float vc = srt[min(a, NC - 1)];
  vc = (a < NC) ? vc : INF;
  float vf = sf[min(b, NF - 1)];
  vf = (b < NF) ? vf : INF;

  float mz[6];
#pragma unroll
  for (int m = 0; m < 6; ++m) {
    bool takeC = vc <= vf;
    mz[m] = takeC ? vc : vf;
    a += takeC ? 1 : 0;
    b += takeC ? 0 : 1;
    int idx = takeC ? min(a, NC - 1) : (NC + min(b, NF - 1));  // single gather
    float nv = srt[idx];
    float nvc = (a < NC) ? nv : INF;
    float nvf = (b < NF) ? nv : INF;
    vc = takeC ? nvc : vc;
    vf = takeC ? vf : nvf;
  }

  // ---- direct coalesced stores; points recomputed from sorted z ----
  float* zo = out_z + (size_t)ray * NT + dg;
  float* po = out_pts + ((size_t)ray * NT + dg) * 3;
#pragma unroll
  for (int m = 0; m < 6; ++m) {
    float zz = mz[m];
    zo[m] = zz;
    po[3 * m + 0] = fmaf(dx, zz, ox);
    po[3 * m + 1] = fmaf(dy, zz, oy);
    po[3 * m + 2] = fmaf(dz, zz, oz);
  }
}

extern "C" void kernel_launch(void* const* d_in, const int* in_sizes, int n_in,
                              void* d_out, int out_size, void* d_ws, size_t ws_size,
                              hipStream_t stream) {
  const float* origins = (const float*)d_in[0];
  const float* dirs = (const float*)d_in[1];
  const float* density = (const float*)d_in[2];
  const float* trand = (const float*)d_in[3];
  const float* u = (const float*)d_in[4];
  const int B = in_sizes[0] / 3;
  float* out_pts = (float*)d_out;                        // [B,192,3]
  float* out_z = (float*)d_out + (size_t)B * NT * 3;     // [B,192]
  const int blocks = (B + WPB - 1) / WPB;
  hipLaunchKernelGGL(hier_sampler_kernel, dim3(blocks), dim3(THREADS), 0, stream,
                     origins, dirs, density, trand, u, out_pts, out_z, B);
}